// MultiHeadAttention_65377992180314
// MI455X (gfx1250) — compile-verified
//
#include <hip/hip_runtime.h>
#include <math.h>

typedef __attribute__((ext_vector_type(2))) float v2f;
typedef __attribute__((ext_vector_type(8))) float v8f;

#define D_MODEL 2048
#define S_LEN   2048
#define N_HEADS 16
#define N_KV    4
#define D_K     128
#define LN_ROPE 9.210340371976184f /* ln(10000) */

__device__ __forceinline__ v8f wmma_f32(v2f a, v2f b, v8f c) {
  // D(16x16,f32) = A(16x4,f32) x B(4x16,f32) + C   -> v_wmma_f32_16x16x4_f32
  return __builtin_amdgcn_wmma_f32_16x16x4_f32(false, a, false, b, (short)0, c,
                                               false, false);
}

// Async global->LDS copy, 16B per lane, tracked by ASYNCcnt (CDNA5 TDM-lite
// path: no VGPR round trip, no ds_store).  LDS destination address is the
// low 32 bits of the generic pointer (flat aperture rule: addr[31:0] is the
// wave-relative LDS offset).
__device__ __forceinline__ uint32_t lds_off32(const void* p) {
  return (uint32_t)(uintptr_t)p;
}
__device__ __forceinline__ void g2l_async_b128(uint32_t lds_dst,
                                               const void* gsrc) {
  asm volatile("global_load_async_to_lds_b128 %0, %1, off"
               :
               : "v"(lds_dst), "v"((unsigned long long)(uintptr_t)gsrc)
               : "memory");
}
__device__ __forceinline__ void wait_async0() {
  asm volatile("s_wait_asynccnt 0" ::: "memory");
}

// ---------------------------------------------------------------------------
// Kernel 1: fused QKV projection + bias + per-head RMSNorm + RoPE.
// Workgroup = 128 threads (4 waves). Tile = 64 rows x 128 cols (one head).
// blockIdx.y: 0..15 -> Q head, 16..19 -> K head, 20..23 -> V head.
// ---------------------------------------------------------------------------
__global__ __launch_bounds__(128)
void qkv_kernel(const float* __restrict__ x,
                const float* __restrict__ Wq, const float* __restrict__ bq,
                const float* __restrict__ Wk, const float* __restrict__ bk,
                const float* __restrict__ Wv, const float* __restrict__ bv,
                const float* __restrict__ qnw, const float* __restrict__ knw,
                float* __restrict__ qb, float* __restrict__ kb,
                float* __restrict__ vb) {
  __shared__ float sA[64 * 36];    // X tile, padded rows (float4-aligned)
  __shared__ float sB[128 * 36];   // W tile
  __shared__ float sC[64 * 132];   // fp32 result tile for row-wise epilogue

  const int t = threadIdx.x;
  const int wav = t >> 5;
  const int lane = t & 31;
  const int mloc = lane & 15;
  const int khalf = (lane >> 4) * 2;
  const int hi = lane >> 4;
  const int m0 = blockIdx.x * 64;
  const int seg = blockIdx.y;

  const float* W;
  const float* bias;
  const float* nw = qnw;
  bool donorm, dorope;
  if (seg < 16) {
    W = Wq + (size_t)seg * 128 * D_MODEL; bias = bq + seg * 128;
    donorm = true; dorope = true; nw = qnw;
  } else if (seg < 20) {
    W = Wk + (size_t)(seg - 16) * 128 * D_MODEL; bias = bk + (seg - 16) * 128;
    donorm = true; dorope = true; nw = knw;
  } else {
    W = Wv + (size_t)(seg - 20) * 128 * D_MODEL; bias = bv + (seg - 20) * 128;
    donorm = false; dorope = false;
  }

  const uint32_t sAo = lds_off32(&sA[0]);
  const uint32_t sBo = lds_off32(&sB[0]);

  const v8f vzero = {0.f, 0.f, 0.f, 0.f, 0.f, 0.f, 0.f, 0.f};
  v8f acc[8];
#pragma unroll
  for (int i = 0; i < 8; ++i) acc[i] = vzero;

  for (int k0 = 0; k0 < D_MODEL; k0 += 32) {
    __syncthreads();
#pragma unroll
    for (int j = 0; j < 4; ++j) {  // stage A: 64x32 (async, 16B/lane)
      int idx = t + j * 128;
      int r = idx >> 3, c4 = (idx & 7) * 4;
      g2l_async_b128(sAo + (uint32_t)(r * 36 + c4) * 4u,
                     &x[(size_t)(m0 + r) * D_MODEL + k0 + c4]);
    }
#pragma unroll
    for (int j = 0; j < 8; ++j) {  // stage B: 128x32 (rows of W, contiguous K)
      int idx = t + j * 128;
      int r = idx >> 3, c4 = (idx & 7) * 4;
      g2l_async_b128(sBo + (uint32_t)(r * 36 + c4) * 4u,
                     &W[(size_t)r * D_MODEL + k0 + c4]);
    }
    wait_async0();
    __syncthreads();
#pragma unroll
    for (int kk = 0; kk < 32; kk += 4) {
      v2f a = *(const v2f*)&sA[(wav * 16 + mloc) * 36 + kk + khalf];
#pragma unroll
      for (int tt = 0; tt < 8; ++tt) {
        v2f bf = *(const v2f*)&sB[(tt * 16 + mloc) * 36 + kk + khalf];
        acc[tt] = wmma_f32(a, bf, acc[tt]);
      }
    }
  }

#pragma unroll
  for (int tt = 0; tt < 8; ++tt)
#pragma unroll
    for (int r = 0; r < 8; ++r)
      sC[(wav * 16 + r + 8 * hi) * 132 + tt * 16 + mloc] = acc[tt][r];
  __syncthreads();

  if (t < 64) {  // one thread per output row: bias + RMSNorm + RoPE + store
    const int row_g = m0 + t;
    const int b = row_g >> 11;
    const int s = row_g & 2047;
    float* dst;
    if (seg < 16)
      dst = qb + ((size_t)(b * N_HEADS + seg) * S_LEN + s) * D_K;
    else if (seg < 20)
      dst = kb + ((size_t)(b * N_KV + (seg - 16)) * S_LEN + s) * D_K;
    else
      dst = vb + ((size_t)(b * N_KV + (seg - 20)) * S_LEN + s) * D_K;

    float rinv = 1.0f;
    if (donorm) {
      float ss = 0.f;
      for (int c = 0; c < 128; ++c) {
        float v = sC[t * 132 + c] + bias[c];
        ss += v * v;
      }
      rinv = 1.0f / sqrtf(ss * (1.0f / 128.0f) + 1e-8f);
    }
    for (int j = 0; j < 64; ++j) {
      float v1 = sC[t * 132 + 2 * j] + bias[2 * j];
      float v2 = sC[t * 132 + 2 * j + 1] + bias[2 * j + 1];
      if (donorm) {
        v1 = v1 * rinv * nw[2 * j];
        v2 = v2 * rinv * nw[2 * j + 1];
      }
      if (dorope) {
        float ang = (float)s * expf(-((float)j * (LN_ROPE / 64.0f)));
        float c = cosf(ang), sn = sinf(ang);
        float o1 = v1 * c - v2 * sn;
        float o2 = v1 * sn + v2 * c;
        v1 = o1; v2 = o2;
      }
      dst[2 * j] = v1;
      dst[2 * j + 1] = v2;
    }
  }
}

// ---------------------------------------------------------------------------
// Kernel 2: causal flash attention, one (b,h, 64-query tile) per workgroup.
// Each wave owns 16 query rows; K-blocks of 16 keys stream through LDS.
// ---------------------------------------------------------------------------
__global__ __launch_bounds__(128)
void attn_kernel(const float* __restrict__ qb, const float* __restrict__ kb,
                 const float* __restrict__ vb, float* __restrict__ ab) {
  __shared__ float sQ[64 * 132];
  __shared__ float sK[16 * 132];
  __shared__ float sVt[128 * 18];  // V transposed: [d][key]
  __shared__ float sP[4 * 16 * 20];

  const int t = threadIdx.x, wav = t >> 5, lane = t & 31;
  const int mloc = lane & 15, khalf = (lane >> 4) * 2, hi = lane >> 4;
  const int m0 = blockIdx.x * 64;
  const int b = blockIdx.y >> 4;
  const int h = blockIdx.y & 15;
  const int kv = h >> 2;
  const float scale = 0.08838834764831845f;  // 1/sqrt(128)

  const float* Qg = qb + ((size_t)(b * N_HEADS + h) * S_LEN) * D_K;
  const float* Kg = kb + ((size_t)(b * N_KV + kv) * S_LEN) * D_K;
  const float* Vg = vb + ((size_t)(b * N_KV + kv) * S_LEN) * D_K;

  const uint32_t sQo = lds_off32(&sQ[0]);
  const uint32_t sKo = lds_off32(&sK[0]);

#pragma unroll
  for (int j = 0; j < 16; ++j) {  // stage Q tile 64x128 (async)
    int idx = t + j * 128;
    int r = idx >> 5, c4 = (idx & 31) * 4;
    g2l_async_b128(sQo + (uint32_t)(r * 132 + c4) * 4u,
                   &Qg[(size_t)(m0 + r) * D_K + c4]);
  }

  const v8f vzero = {0.f, 0.f, 0.f, 0.f, 0.f, 0.f, 0.f, 0.f};
  v8f oacc[8];
#pragma unroll
  for (int i = 0; i < 8; ++i) oacc[i] = vzero;
  float mrow[8], lrow[8];
#pragma unroll
  for (int r = 0; r < 8; ++r) {
    mrow[r] = -1e30f;
    lrow[r] = 0.f;
  }

  const int q0w = m0 + wav * 16;
  const int nblk = (m0 >> 4) + 4;  // uniform causal bound for whole workgroup

  for (int blk = 0; blk < nblk; ++blk) {
    const int kb0 = blk * 16;
    __syncthreads();
#pragma unroll
    for (int j = 0; j < 4; ++j) {  // stage K async; V transposed manually
      int idx = t + j * 128;
      int r = idx >> 5, c4 = (idx & 31) * 4;
      g2l_async_b128(sKo + (uint32_t)(r * 132 + c4) * 4u,
                     &Kg[(size_t)(kb0 + r) * D_K + c4]);
      float4 v4 = *(const float4*)&Vg[(size_t)(kb0 + r) * D_K + c4];
      sVt[(c4 + 0) * 18 + r] = v4.x;
      sVt[(c4 + 1) * 18 + r] = v4.y;
      sVt[(c4 + 2) * 18 + r] = v4.z;
      sVt[(c4 + 3) * 18 + r] = v4.w;
    }
    wait_async0();  // covers Q stage on first iteration, K stage on all
    __syncthreads();

    v8f sc = vzero;  // S = Q Kᵀ for this 16x16 tile
#pragma unroll
    for (int kd = 0; kd < 128; kd += 4) {
      v2f a = *(const v2f*)&sQ[(wav * 16 + mloc) * 132 + kd + khalf];
      v2f bf = *(const v2f*)&sK[mloc * 132 + kd + khalf];
      sc = wmma_f32(a, bf, sc);
    }

#pragma unroll
    for (int r = 0; r < 8; ++r) {  // online softmax per row
      const int qg = q0w + r + 8 * hi;
      const int kg = kb0 + mloc;
      float sv = sc[r] * scale;
      if (kg > qg) sv = -1e9f;
      float mx = sv;
#pragma unroll
      for (int off = 1; off < 16; off <<= 1)
        mx = fmaxf(mx, __shfl_xor(mx, off, 32));
      float mnew = fmaxf(mrow[r], mx);
      float p = expf(sv - mnew);
      float ps = p;
#pragma unroll
      for (int off = 1; off < 16; off <<= 1) ps += __shfl_xor(ps, off, 32);
      float corr = expf(mrow[r] - mnew);
      lrow[r] = lrow[r] * corr + ps;
      mrow[r] = mnew;
#pragma unroll
      for (int tt = 0; tt < 8; ++tt) oacc[tt][r] *= corr;
      sP[wav * 320 + (r + 8 * hi) * 20 + mloc] = p;
    }
    __syncthreads();

#pragma unroll
    for (int tt = 0; tt < 8; ++tt) {  // O += P V
#pragma unroll
      for (int kk = 0; kk < 16; kk += 4) {
        v2f a = *(const v2f*)&sP[wav * 320 + mloc * 20 + kk + khalf];
        v2f bf = *(const v2f*)&sVt[(tt * 16 + mloc) * 18 + kk + khalf];
        oacc[tt] = wmma_f32(a, bf, oacc[tt]);
      }
    }
  }

#pragma unroll
  for (int r = 0; r < 8; ++r) {  // normalize, write [B,S,H*D_K]
    const int qrow = q0w + r + 8 * hi;
    const float inv = 1.0f / lrow[r];
    const size_t base = ((size_t)b * S_LEN + qrow) * D_MODEL + h * D_K;
#pragma unroll
    for (int tt = 0; tt < 8; ++tt)
      ab[base + tt * 16 + mloc] = oacc[tt][r] * inv;
  }
}

// ---------------------------------------------------------------------------
// Kernel 3: output projection  out = attn · Woᵀ + bo
// ---------------------------------------------------------------------------
__global__ __launch_bounds__(128)
void oproj_kernel(const float* __restrict__ ain, const float* __restrict__ Wo,
                  const float* __restrict__ bo, float* __restrict__ out) {
  __shared__ float sA[64 * 36];
  __shared__ float sB[128 * 36];
  const int t = threadIdx.x, wav = t >> 5, lane = t & 31;
  const int mloc = lane & 15, khalf = (lane >> 4) * 2, hi = lane >> 4;
  const int m0 = blockIdx.x * 64;
  const int n0 = blockIdx.y * 128;

  const uint32_t sAo = lds_off32(&sA[0]);
  const uint32_t sBo = lds_off32(&sB[0]);

  const v8f vzero = {0.f, 0.f, 0.f, 0.f, 0.f, 0.f, 0.f, 0.f};
  v8f acc[8];
#pragma unroll
  for (int i = 0; i < 8; ++i) acc[i] = vzero;

  for (int k0 = 0; k0 < D_MODEL; k0 += 32) {
    __syncthreads();
#pragma unroll
    for (int j = 0; j < 4; ++j) {
      int idx = t + j * 128;
      int r = idx >> 3, c4 = (idx & 7) * 4;
      g2l_async_b128(sAo + (uint32_t)(r * 36 + c4) * 4u,
                     &ain[(size_t)(m0 + r) * D_MODEL + k0 + c4]);
    }
#pragma unroll
    for (int j = 0; j < 8; ++j) {
      int idx = t + j * 128;
      int r = idx >> 3, c4 = (idx & 7) * 4;
      g2l_async_b128(sBo + (uint32_t)(r * 36 + c4) * 4u,
                     &Wo[(size_t)(n0 + r) * D_MODEL + k0 + c4]);
    }
    wait_async0();
    __syncthreads();
#pragma unroll
    for (int kk = 0; kk < 32; kk += 4) {
      v2f a = *(const v2f*)&sA[(wav * 16 + mloc) * 36 + kk + khalf];
#pragma unroll
      for (int tt = 0; tt < 8; ++tt) {
        v2f bf = *(const v2f*)&sB[(tt * 16 + mloc) * 36 + kk + khalf];
        acc[tt] = wmma_f32(a, bf, acc[tt]);
      }
    }
  }

#pragma unroll
  for (int tt = 0; tt < 8; ++tt) {
    float bvv = bo[n0 + tt * 16 + mloc];
#pragma unroll
    for (int r = 0; r < 8; ++r) {
      int row = m0 + wav * 16 + r + 8 * hi;
      out[(size_t)row * D_MODEL + n0 + tt * 16 + mloc] = acc[tt][r] + bvv;
    }
  }
}

extern "C" void kernel_launch(void* const* d_in, const int* in_sizes, int n_in,
                              void* d_out, int out_size, void* d_ws,
                              size_t ws_size, hipStream_t stream) {
  const float* x = (const float*)d_in[0];
  const float* Wq = (const float*)d_in[1];
  const float* bq = (const float*)d_in[2];
  const float* Wk = (const float*)d_in[3];
  const float* bk = (const float*)d_in[4];
  const float* Wv = (const float*)d_in[5];
  const float* bv = (const float*)d_in[6];
  const float* Wo = (const float*)d_in[7];
  const float* bo = (const float*)d_in[8];
  const float* qnw = (const float*)d_in[9];
  const float* knw = (const float*)d_in[10];
  float* out = (float*)d_out;

  float* ws = (float*)d_ws;
  float* qbuf = ws;                                        // [2,16,2048,128]
  float* kbuf = qbuf + (size_t)2 * N_HEADS * S_LEN * D_K;  // [2,4,2048,128]
  float* vbuf = kbuf + (size_t)2 * N_KV * S_LEN * D_K;     // [2,4,2048,128]
  float* abuf = vbuf + (size_t)2 * N_KV * S_LEN * D_K;     // [4096,2048]

  qkv_kernel<<<dim3(64, 24), 128, 0, stream>>>(x, Wq, bq, Wk, bk, Wv, bv, qnw,
                                               knw, qbuf, kbuf, vbuf);
  attn_kernel<<<dim3(32, 32), 128, 0, stream>>>(qbuf, kbuf, vbuf, abuf);
  oproj_kernel<<<dim3(64, 16), 128, 0, stream>>>(abuf, Wo, bo, out);
}